// GaussianRasterizer_39101382263436
// MI455X (gfx1250) — compile-verified
//
#include <hip/hip_runtime.h>
#include <hip/hip_bf16.h>

#define HH 256
#define WW 256
#define FXc 256.0f
#define FYc 256.0f
#define NEARc 0.2f
#define BLURc 0.3f
#define AMIN (1.0f/255.0f)
#define AMAX 0.99f
#define NG 8192
#define CHG 64                 // gaussians per LDS chunk
#define HWSZ (HH*WW)

typedef __attribute__((ext_vector_type(2))) float v2f;
typedef __attribute__((ext_vector_type(8))) float v8f;
typedef __attribute__((ext_vector_type(4))) int   v4i;

#if defined(__HIP_DEVICE_COMPILE__) && defined(__gfx1250__) && \
    __has_builtin(__builtin_amdgcn_global_load_async_to_lds_b128)
#define HAVE_ASYNC 1
#else
#define HAVE_ASYNC 0
#endif

// ---------------- preprocess: per-gaussian conic / projection ----------------
__global__ void __launch_bounds__(256) k_preprocess(
    const float* __restrict__ means3D, const float* __restrict__ opac,
    const float* __restrict__ cols, const float* __restrict__ scales,
    const float* __restrict__ rots, float* __restrict__ depth,
    float4* __restrict__ G, int* __restrict__ radii) {
  int i = blockIdx.x * 256 + threadIdx.x;
  if (i >= NG) return;
  float tx = means3D[3*i+0], ty = means3D[3*i+1], tz = means3D[3*i+2];
  float qr = rots[4*i+0], qx = rots[4*i+1], qy = rots[4*i+2], qz = rots[4*i+3];
  float qn = rsqrtf(qr*qr + qx*qx + qy*qy + qz*qz);
  qr *= qn; qx *= qn; qy *= qn; qz *= qn;
  float R00 = 1.f-2.f*(qy*qy+qz*qz), R01 = 2.f*(qx*qy-qr*qz), R02 = 2.f*(qx*qz+qr*qy);
  float R10 = 2.f*(qx*qy+qr*qz), R11 = 1.f-2.f*(qx*qx+qz*qz), R12 = 2.f*(qy*qz-qr*qx);
  float R20 = 2.f*(qx*qz-qr*qy), R21 = 2.f*(qy*qz+qr*qx), R22 = 1.f-2.f*(qx*qx+qy*qy);
  float sx = scales[3*i+0], sy = scales[3*i+1], sz = scales[3*i+2];
  float M00=R00*sx, M01=R01*sy, M02=R02*sz;
  float M10=R10*sx, M11=R11*sy, M12=R12*sz;
  float M20=R20*sx, M21=R21*sy, M22=R22*sz;
  float c00 = M00*M00 + M01*M01 + M02*M02;
  float c01 = M00*M10 + M01*M11 + M02*M12;
  float c02 = M00*M20 + M01*M21 + M02*M22;
  float c11 = M10*M10 + M11*M11 + M12*M12;
  float c12 = M10*M20 + M11*M21 + M12*M22;
  float c22 = M20*M20 + M21*M21 + M22*M22;
  const float limx = 1.3f*0.5f, limy = 1.3f*0.5f;
  float invz = 1.0f / tz, z2 = tz*tz;
  float txc = fminf(fmaxf(tx*invz, -limx), limx) * tz;
  float tyc = fminf(fmaxf(ty*invz, -limy), limy) * tz;
  float j00 = FXc*invz, j02 = -FXc*txc/z2;
  float j11 = FYc*invz, j12 = -FYc*tyc/z2;
  float u0 = c00*j00 + c02*j02;
  float u1 = c01*j00 + c12*j02;
  float u2 = c02*j00 + c22*j02;
  float A2 = j00*u0 + j02*u2;
  float B2 = j11*u1 + j12*u2;
  float v1v = c11*j11 + c12*j12;
  float v2v = c12*j11 + c22*j12;
  float D2 = j11*v1v + j12*v2v;
  float a = A2 + BLURc, b = B2, d = D2 + BLURc;
  float det = a*d - b*b;
  bool valid = (tz > NEARc) && (det > 0.f);
  float dets = (det != 0.f) ? det : 1.f;
  float ca = d/dets, cb = -b/dets, cc = a/dets;
  float mid = 0.5f*(a+d);
  float lam1 = mid + sqrtf(fmaxf(0.1f, mid*mid - det));
  radii[i] = valid ? (int)ceilf(3.0f*sqrtf(lam1)) : 0;
  float px = FXc*tx*invz + 0.5f*(WW-1);
  float py = FYc*ty*invz + 0.5f*(HH-1);
  float opv = valid ? opac[i] : 0.f;
  float4 f0 = make_float4(px, py, ca, cb);
  float4 f1 = make_float4(cc, opv, invz, 0.f);
  float4 f2 = make_float4(cols[3*i+0], cols[3*i+1], cols[3*i+2], invz);
  float4 f3 = make_float4(0.f, 0.f, 0.f, 0.f);
  G[4*i+0] = f0; G[4*i+1] = f1; G[4*i+2] = f2; G[4*i+3] = f3;
  depth[i] = tz;
}

// ---------------- single-workgroup LDS bitonic argsort (8192 keys) ----------
__global__ void __launch_bounds__(1024) k_sort(const float* __restrict__ depth,
                                               int* __restrict__ sidx) {
  __shared__ float sk[NG];
  __shared__ int   si[NG];
  int t = threadIdx.x;
  for (int i = t; i < NG; i += 1024) { sk[i] = depth[i]; si[i] = i; }
  __syncthreads();
  for (int k = 2; k <= NG; k <<= 1) {
    for (int j = k >> 1; j > 0; j >>= 1) {
      for (int i = t; i < NG; i += 1024) {
        int ixj = i ^ j;
        if (ixj > i) {
          float a = sk[i], b2 = sk[ixj];
          bool up = ((i & k) == 0);
          bool sw = up ? (a > b2) : (a < b2);
          if (sw) {
            sk[i] = b2; sk[ixj] = a;
            int tmp = si[i]; si[i] = si[ixj]; si[ixj] = tmp;
          }
        }
      }
      __syncthreads();
    }
  }
  for (int i = t; i < NG; i += 1024) sidx[i] = si[i];
}

// ---------------- gather sorted packed table --------------------------------
__global__ void __launch_bounds__(256) k_gather(const int* __restrict__ sidx,
                                                const float4* __restrict__ G,
                                                float4* __restrict__ Gs) {
  int i = blockIdx.x * 256 + threadIdx.x;
  int j = sidx[i];
#pragma unroll
  for (int t = 0; t < 4; ++t) Gs[4*i+t] = G[4*j+t];
}

// ---------------- rasterizer helpers ----------------------------------------
__device__ __forceinline__ void issue_copy(const float4* __restrict__ src,
                                           float4* dst, int tid) {
#if HAVE_ASYNC
  typedef __attribute__((address_space(1))) v4i* gp_t;
  typedef __attribute__((address_space(3))) v4i* lp_t;
  __builtin_amdgcn_global_load_async_to_lds_b128(
      (gp_t)(float4*)(src + tid), (lp_t)(dst + tid), 0, 0);
#else
  dst[tid] = src[tid];
#endif
}

__device__ __forceinline__ void wait_async0() {
#if HAVE_ASYNC
#if __has_builtin(__builtin_amdgcn_s_wait_asynccnt)
  __builtin_amdgcn_s_wait_asynccnt(0);
#else
  asm volatile("s_wait_asynccnt 0" ::: "memory");
#endif
#endif
}

// ---------------- tiled rasterizer: 16x8 pixels/block, WMMA accumulate ------
__global__ void __launch_bounds__(256) k_raster(const float4* __restrict__ Gs,
                                                const float* __restrict__ bg,
                                                float* __restrict__ out) {
  __shared__ float4 sbuf[2][CHG * 4];
  int tid  = threadIdx.x;
  int lane = tid & 31;
  int wave = tid >> 5;
  int lc = lane & 15;        // pixel column within tile (both halves share it)
  int h  = lane >> 4;        // lane-half: K pair 0/1 vs 2/3
  int ccl = lc & 3;          // clamped channel index for A-matrix reads
  float am = (lc < 4) ? 1.f : 0.f;  // channel-valid mask (rows 4..15 are zero)
  int x = blockIdx.x * 16 + lc;
  int y = blockIdx.y * 8 + wave;
  float xf = (float)x, yf = (float)y;
  float bg0 = bg[0], bg1 = bg[1], bg2 = bg[2];

  v8f acc = {0.f,0.f,0.f,0.f,0.f,0.f,0.f,0.f};
  float T = 1.0f;

  issue_copy(Gs, &sbuf[0][0], tid);                 // prologue: chunk 0
  const int NCH = NG / CHG;
  for (int c = 0; c < NCH; ++c) {
    wait_async0();
    __syncthreads();
    if (c + 1 < NCH)
      issue_copy(Gs + (size_t)(c + 1) * CHG * 4, &sbuf[(c + 1) & 1][0], tid);
    const float4* buf = &sbuf[c & 1][0];

    bool waveDone = __all(T < 1e-4f);               // wave-uniform skip
    if (!waveDone) {
#pragma unroll 4
      for (int s = 0; s < CHG; s += 4) {
        const float4* gE = &buf[(s + 2*h + 0) * 4]; // "even" K of this half
        const float4* gO = &buf[(s + 2*h + 1) * 4]; // "odd"  K of this half
        float4 Pe = gE[0], Qe = gE[1];
        float4 Po = gO[0], Qo = gO[1];

        float dxe = xf - Pe.x, dye = yf - Pe.y;
        float pwe = -0.5f*(Pe.z*dxe*dxe + Qe.x*dye*dye) - Pe.w*dxe*dye;
        float ae = fminf(Qe.y * __expf(pwe), AMAX);
        ae = (pwe <= 0.f && ae >= AMIN) ? ae : 0.f;

        float dxo = xf - Po.x, dyo = yf - Po.y;
        float pwo = -0.5f*(Po.z*dxo*dxo + Qo.x*dyo*dyo) - Po.w*dxo*dyo;
        float ao = fminf(Qo.y * __expf(pwo), AMAX);
        ao = (pwo <= 0.f && ao >= AMIN) ? ao : 0.f;

        float Pm = (1.f - ae) * (1.f - ao);         // my half's (1-a)(1-a)
        float Pt = __shfl_xor(Pm, 16, 32);          // other half's product
        float base = h ? (T * Pt) : T;              // half1 starts after K0,K1
        float we = base * ae;
        float wo = base * (1.f - ae) * ao;
        T = T * Pm * Pt;

        // A = colors (16 channels x 4 K): rows 0..3 = r,g,b,invdepth.
        // Unconditional clamped-index load + mask multiply: no exec divergence.
        const float* ce = (const float*)&gE[2];
        const float* co = (const float*)&gO[2];
        v2f A; A.x = ce[ccl] * am;
               A.y = co[ccl] * am;
        v2f B; B.x = we; B.y = wo;                  // B = weights (4 K x 16 px)
        acc = __builtin_amdgcn_wmma_f32_16x16x4_f32(
            false, A, false, B, (short)0, acc, false, false);
      }
    }
  }

  if (lane < 16) {                                  // rows 0..3 live in half 0
    int p = y * WW + x;
    out[0*HWSZ + p] = acc[0] + T * bg0;
    out[1*HWSZ + p] = acc[1] + T * bg1;
    out[2*HWSZ + p] = acc[2] + T * bg2;
    out[3*HWSZ + NG + p] = acc[3];                  // invdepth plane after radii
  }
}

// ---------------- host-side launcher ----------------------------------------
extern "C" void kernel_launch(void* const* d_in, const int* in_sizes, int n_in,
                              void* d_out, int out_size, void* d_ws, size_t ws_size,
                              hipStream_t stream) {
  const float* means3D = (const float*)d_in[0];
  // d_in[1] = means2D (unused by reference math)
  const float* opac   = (const float*)d_in[2];
  const float* cols   = (const float*)d_in[3];
  const float* scales = (const float*)d_in[4];
  const float* rots   = (const float*)d_in[5];
  const float* bg     = (const float*)d_in[6];
  float* out = (float*)d_out;
  int* radii = (int*)d_out + 3 * HWSZ;

  float* depth = (float*)d_ws;                      // 8192 f32
  int*   sidx  = (int*)d_ws + NG;                   // 8192 i32
  float4* G    = (float4*)((char*)d_ws + 64 * 1024);// 512 KB unsorted
  float4* Gsrt = G + (size_t)NG * 4;                // 512 KB sorted

  k_preprocess<<<NG / 256, 256, 0, stream>>>(means3D, opac, cols, scales, rots,
                                             depth, G, radii);
  k_sort<<<1, 1024, 0, stream>>>(depth, sidx);
  k_gather<<<NG / 256, 256, 0, stream>>>(sidx, G, Gsrt);
  k_raster<<<dim3(WW / 16, HH / 8), 256, 0, stream>>>(Gsrt, bg, out);
}